// GATv3_block_14388140442032
// MI455X (gfx1250) — compile-verified
//
#include <hip/hip_runtime.h>
#include <hip/hip_bf16.h>

typedef __attribute__((ext_vector_type(2))) float v2f;
typedef __attribute__((ext_vector_type(8))) float v8f;

#define NEG_SLOPE 0.2f

// ---------- helpers ----------

__device__ __forceinline__ unsigned encodeOrd(float f) {
    unsigned u = __float_as_uint(f);
    return (u & 0x80000000u) ? ~u : (u | 0x80000000u);
}
__device__ __forceinline__ float decodeOrd(unsigned u) {
    unsigned v = (u & 0x80000000u) ? (u & 0x7fffffffu) : ~u;
    return __uint_as_float(v);
}

// One K=4 step of a 16x16 fp32 WMMA. Layout per ISA 7.12.2:
//   A 16x4: lanes0-15 rows 0-15 {K=k,k+1}; lanes16-31 rows 0-15 {K=k+2,k+3}
//   B 4x16: VGPR0 = row k (lanes0-15) / row k+2 (lanes16-31); VGPR1 = k+1 / k+3
__device__ __forceinline__ v8f wmma_step_f32(v8f acc,
                                             const float* __restrict__ A, int arow, int strideA,
                                             const float* __restrict__ W, int strideW,
                                             int k, int col0, int lo, int hi) {
    int ka = k + 2 * hi;
    v2f a, b;
    a.x = A[arow * strideA + ka];
    a.y = A[arow * strideA + ka + 1];
    b.x = W[ka * strideW + col0 + lo];
    b.y = W[(ka + 1) * strideW + col0 + lo];
    return __builtin_amdgcn_wmma_f32_16x16x4_f32(false, a, false, b, (short)0, acc,
                                                 false, false);
}

// ---------- kernels ----------

// workspace init
__global__ void init_kernel(float* __restrict__ accum, float* __restrict__ den,
                            unsigned* __restrict__ mx, float* __restrict__ meansum,
                            float* __restrict__ cnt,
                            int nAcc, int nDen, int nMean, int nCnt) {
    int t = blockIdx.x * blockDim.x + threadIdx.x;
    if (t < nAcc) accum[t] = 0.0f;
    if (t < nDen) { den[t] = 0.0f; mx[t] = encodeOrd(-__builtin_inff()); }
    if (t < nMean) meansum[t] = 0.0f;
    if (t < nCnt) cnt[t] = 0.0f;
}

// generic C[M,Nc] = A[M,K] @ W[K,Nc] (+bias) (+relu), fp32 WMMA, one wave per 16x16 tile
__global__ void wmma_gemm_f32(const float* __restrict__ A, const float* __restrict__ W,
                              const float* __restrict__ bias, float* __restrict__ C,
                              int M, int K, int Nc, int doRelu) {
    int lane = threadIdx.x;
    int lo = lane & 15, hi = lane >> 4;
    int row0 = blockIdx.x * 16;
    int col0 = blockIdx.y * 16;
    int arow = row0 + lo; if (arow >= M) arow = M - 1;
    v8f acc = {};
    for (int k = 0; k < K; k += 4)
        acc = wmma_step_f32(acc, A, arow, K, W, Nc, k, col0, lo, hi);
    int col = col0 + lo;
    float bv = bias ? bias[col] : 0.0f;
#pragma unroll
    for (int r = 0; r < 8; ++r) {
        int row = row0 + r + 8 * hi;
        if (row < M) {
            float v = acc[r] + bv;
            if (doRelu) v = fmaxf(v, 0.0f);
            C[row * Nc + col] = v;
        }
    }
}

// e = relu(edge_attr @ W_edge[64:96,:] + px[src] + b_edge), 16 edges x 16 cols per wave
__global__ void edge_e_kernel(const float* __restrict__ ea, const float* __restrict__ Wb,
                              const float* __restrict__ b_edge, const float* __restrict__ px,
                              const int* __restrict__ srcI, float* __restrict__ e_out, int E) {
    int lane = threadIdx.x;
    int lo = lane & 15, hi = lane >> 4;
    int row0 = blockIdx.x * 16;
    int col0 = blockIdx.y * 16;
    int arow = row0 + lo; if (arow >= E) arow = E - 1;
    v8f acc = {};
    for (int k = 0; k < 32; k += 4)
        acc = wmma_step_f32(acc, ea, arow, 32, Wb, 32, k, col0, lo, hi);
    int col = col0 + lo;
    float bv = b_edge[col];
#pragma unroll
    for (int r = 0; r < 8; ++r) {
        int row = row0 + r + 8 * hi;
        if (row < E) {
            int s = srcI[row];
            float v = acc[r] + px[s * 32 + col] + bv;
            e_out[row * 32 + col] = fmaxf(v, 0.0f);
        }
    }
}

// fused: ew = e @ W_e (16x96 tile via WMMA, staged in LDS), then
// logit[e,h] = sum_c lrelu(ew + xl[src] + xr[dst]) * att[h,c]; atomic segment max
__global__ void edge_logits_kernel(const float* __restrict__ e_mat, const float* __restrict__ W_e,
                                   const float* __restrict__ xl, const float* __restrict__ xr,
                                   const float* __restrict__ att,
                                   const int* __restrict__ srcI, const int* __restrict__ dstI,
                                   float* __restrict__ logits, unsigned* __restrict__ mx, int E) {
    __shared__ float lds[16 * 96];
    int lane = threadIdx.x;
    int lo = lane & 15, hi = lane >> 4;
    int e0 = blockIdx.x * 16;
    int arow = e0 + lo; if (arow >= E) arow = E - 1;
#pragma unroll
    for (int ct = 0; ct < 6; ++ct) {
        v8f acc = {};
        for (int k = 0; k < 32; k += 4)
            acc = wmma_step_f32(acc, e_mat, arow, 32, W_e, 96, k, ct * 16, lo, hi);
#pragma unroll
        for (int r = 0; r < 8; ++r)
            lds[(r + 8 * hi) * 96 + ct * 16 + lo] = acc[r];
    }
    __syncthreads();
    // lane == channel c; loop over the 48 (edge,head) pairs of this tile
    for (int p = 0; p < 48; ++p) {
        int i = p / 3, h = p - 3 * i;
        int row = e0 + i;
        if (row >= E) break;
        int s = srcI[row], d = dstI[row];
        float v = lds[i * 96 + h * 32 + lane] + xl[s * 96 + h * 32 + lane]
                + xr[d * 96 + h * 32 + lane];
        float lr = (v > 0.0f) ? v : NEG_SLOPE * v;
        float t = lr * att[h * 32 + lane];
#pragma unroll
        for (int off = 16; off > 0; off >>= 1) t += __shfl_xor(t, off, 32);
        if (lane == 0) {
            logits[row * 3 + h] = t;
            atomicMax(&mx[d * 3 + h], encodeOrd(t));
        }
    }
}

// wave per edge: ex = exp(logit - mx[dst]); den[dst,h] += ex; acc[dst,h,c] += ex*xl[src,h,c]
__global__ void edge_accum_kernel(const float* __restrict__ logits, const unsigned* __restrict__ mx,
                                  const float* __restrict__ xl,
                                  const int* __restrict__ srcI, const int* __restrict__ dstI,
                                  float* __restrict__ den, float* __restrict__ accum, int E) {
    int t = blockIdx.x * blockDim.x + threadIdx.x;
    int edge = t >> 5, lane = t & 31;
    if (edge >= E) return;
    int s = srcI[edge], d = dstI[edge];
    float ex[3];
#pragma unroll
    for (int h = 0; h < 3; ++h)
        ex[h] = expf(logits[edge * 3 + h] - decodeOrd(mx[d * 3 + h]));
    if (lane < 3) atomicAdd(&den[d * 3 + lane], ex[lane]);
#pragma unroll
    for (int h = 0; h < 3; ++h)
        atomicAdd(&accum[d * 96 + h * 32 + lane], ex[h] * xl[s * 96 + h * 32 + lane]);
}

// wave per node: gat = acc/den + bias_gat; x_new = relu([gat, glob[batch]] @ W_n2 + b_n2)
// also accumulate scatter-mean numerator + counts
__global__ void node_final_kernel(const float* __restrict__ accum, const float* __restrict__ den,
                                  const float* __restrict__ bias_gat, const float* __restrict__ glob,
                                  const int* __restrict__ batch,
                                  const float* __restrict__ W_n2, const float* __restrict__ b_n2,
                                  float* __restrict__ x_new, float* __restrict__ meansum,
                                  float* __restrict__ cnt, int N) {
    int t = blockIdx.x * blockDim.x + threadIdx.x;
    int n = t >> 5, lane = t & 31;
    if (n >= N) return;
    int b = batch[n];
    float inv[3];
#pragma unroll
    for (int h = 0; h < 3; ++h) {
        float dd = den[n * 3 + h];
        inv[h] = (dd > 0.0f) ? 1.0f / dd : 0.0f;
    }
    float sum = 0.0f;
    for (int k = 0; k < 96; ++k) {
        float a = accum[n * 96 + k] * inv[k >> 5] + bias_gat[k];
        sum = fmaf(a, W_n2[k * 32 + lane], sum);
    }
    for (int k = 0; k < 32; ++k)
        sum = fmaf(glob[b * 32 + k], W_n2[(96 + k) * 32 + lane], sum);
    float v = fmaxf(sum + b_n2[lane], 0.0f);
    x_new[n * 32 + lane] = v;
    atomicAdd(&meansum[b * 32 + lane], v);
    if (lane == 0) atomicAdd(&cnt[b], 1.0f);
}

// per-graph: u_new = relu([glob, mean] @ W_g + b_g)
__global__ void global_final_kernel(const float* __restrict__ glob, const float* __restrict__ meansum,
                                    const float* __restrict__ cnt, const float* __restrict__ W_g,
                                    const float* __restrict__ b_g, float* __restrict__ u_new) {
    int b = blockIdx.x, lane = threadIdx.x;
    float inv = 1.0f / fmaxf(cnt[b], 1.0f);
    float sum = 0.0f;
    for (int k = 0; k < 32; ++k)
        sum = fmaf(glob[b * 32 + k], W_g[k * 32 + lane], sum);
    for (int k = 0; k < 32; ++k)
        sum = fmaf(meansum[b * 32 + k] * inv, W_g[(32 + k) * 32 + lane], sum);
    u_new[b * 32 + lane] = fmaxf(sum + b_g[lane], 0.0f);
}

// ---------- launch ----------

extern "C" void kernel_launch(void* const* d_in, const int* in_sizes, int n_in,
                              void* d_out, int out_size, void* d_ws, size_t ws_size,
                              hipStream_t stream) {
    const float* x         = (const float*)d_in[0];
    const float* edge_attr = (const float*)d_in[1];
    const float* glob      = (const float*)d_in[2];
    const float* W_edge    = (const float*)d_in[3];
    const float* b_edge    = (const float*)d_in[4];
    const float* W_l       = (const float*)d_in[5];
    const float* b_l       = (const float*)d_in[6];
    const float* W_r       = (const float*)d_in[7];
    const float* b_r       = (const float*)d_in[8];
    const float* W_e       = (const float*)d_in[9];
    const float* att       = (const float*)d_in[10];
    const float* bias_gat  = (const float*)d_in[11];
    const float* W_n2      = (const float*)d_in[12];
    const float* b_n2      = (const float*)d_in[13];
    const float* W_g       = (const float*)d_in[14];
    const float* b_g       = (const float*)d_in[15];
    const int* edge_index  = (const int*)d_in[16];
    const int* batch       = (const int*)d_in[17];

    const int N = in_sizes[17];          // 100000
    const int E = in_sizes[16] / 2;      // 1600000
    const int B = in_sizes[2] / 32;      // 64

    const int* srcI = edge_index;
    const int* dstI = edge_index + E;

    // workspace layout (floats)
    float* ws      = (float*)d_ws;
    float* xl      = ws;                       // N*96
    float* xr      = xl + (size_t)N * 96;      // N*96
    float* px      = xr + (size_t)N * 96;      // N*32
    float* logits  = px + (size_t)N * 32;      // E*3
    float* den     = logits + (size_t)E * 3;   // N*3
    float* accum   = den + (size_t)N * 3;      // N*96
    float* meansum = accum + (size_t)N * 96;   // B*32
    float* cnt     = meansum + (size_t)B * 32; // B
    unsigned* mx   = (unsigned*)(cnt + B);     // N*3

    // output layout: x_new [N,32], e [E,32], u_new [B,32]
    float* x_new = (float*)d_out;
    float* e_out = x_new + (size_t)N * 32;
    float* u_new = e_out + (size_t)E * 32;

    int initN = N * 96;
    init_kernel<<<(initN + 255) / 256, 256, 0, stream>>>(accum, den, mx, meansum, cnt,
                                                         N * 96, N * 3, B * 32, B);

    dim3 gN96((N + 15) / 16, 6), gN32((N + 15) / 16, 2), gE32((E + 15) / 16, 2);
    wmma_gemm_f32<<<gN96, 32, 0, stream>>>(x, W_l, b_l, xl, N, 64, 96, 0);
    wmma_gemm_f32<<<gN96, 32, 0, stream>>>(x, W_r, b_r, xr, N, 64, 96, 0);
    wmma_gemm_f32<<<gN32, 32, 0, stream>>>(x, W_edge, nullptr, px, N, 64, 32, 0);

    edge_e_kernel<<<gE32, 32, 0, stream>>>(edge_attr, W_edge + 64 * 32, b_edge, px,
                                           srcI, e_out, E);

    edge_logits_kernel<<<(E + 15) / 16, 32, 0, stream>>>(e_out, W_e, xl, xr, att,
                                                         srcI, dstI, logits, mx, E);

    long long tAcc = (long long)E * 32;
    edge_accum_kernel<<<(int)((tAcc + 255) / 256), 256, 0, stream>>>(logits, mx, xl, srcI, dstI,
                                                                     den, accum, E);

    long long tNode = (long long)N * 32;
    node_final_kernel<<<(int)((tNode + 255) / 256), 256, 0, stream>>>(accum, den, bias_gat, glob,
                                                                      batch, W_n2, b_n2, x_new,
                                                                      meansum, cnt, N);

    global_final_kernel<<<B, 32, 0, stream>>>(glob, meansum, cnt, W_g, b_g, u_new);
}